// LinearFlow_44650480009354
// MI455X (gfx1250) — compile-verified
//
#include <hip/hip_runtime.h>

typedef __attribute__((ext_vector_type(2))) float v2f;
typedef __attribute__((ext_vector_type(4))) float v4f;
typedef __attribute__((ext_vector_type(8))) float v8f;

#define N_ROWS 16384   // x rows
#define N_T    512     // t values
#define LDS_STRIDE 20  // padded row stride (floats) to avoid bank conflicts

// ---------------------------------------------------------------------------
// Phase 1: E[t] = expm(t * A), A = 0.5*((M+M0) - (M+M0)^T), packed directly
// into the per-lane VGPR layout of the WMMA B-fragment (4x16 f32, K-split=2).
//
// Epack[tpair][kstep][lane] : v2f, lane = 16*half + 8*(t&1) + i
//   half selects K0 = 2*half inside the 4-wide K slice, kstep selects j-base
//   4*kstep: value = ( E[t,i, 2*half+4*kstep], E[t,i, 2*half+4*kstep+1] )
// One thread per (t, row i): 8 threads per t, 32 t per 256-thread block.
// ---------------------------------------------------------------------------
__global__ __launch_bounds__(256) void expm_pack_kernel(
    const float* __restrict__ tvec, const float* __restrict__ M,
    const float* __restrict__ M0, v2f* __restrict__ epack)
{
    __shared__ float A[64];
    const int tid = threadIdx.x;
    if (tid < 64) {
        const int i = tid >> 3, j = tid & 7;
        A[tid] = 0.5f * ((M[i * 8 + j] + M0[i * 8 + j]) -
                         (M[j * 8 + i] + M0[j * 8 + i]));
    }
    __syncthreads();

    const int tIdx = blockIdx.x * 32 + (tid >> 3);   // 16 blocks * 32 = 512 t
    const int i    = tid & 7;                        // row of E this thread owns
    const float s  = tvec[tIdx];

    // row = e_i * expm(s*A) via Taylor: term_k = term_{k-1} @ A * (s/k)
    float row[8], term[8], tmp[8];
#pragma unroll
    for (int c = 0; c < 8; ++c) {
        row[c]  = (c == i) ? 1.0f : 0.0f;
        term[c] = row[c];
    }
#pragma unroll
    for (int k = 1; k <= 12; ++k) {
        const float inv = s / (float)k;
#pragma unroll
        for (int c = 0; c < 8; ++c) {
            float acc = 0.0f;
#pragma unroll
            for (int j = 0; j < 8; ++j)
                acc = fmaf(term[j], A[j * 8 + c], acc);
            tmp[c] = acc * inv;
        }
#pragma unroll
        for (int c = 0; c < 8; ++c) {
            term[c] = tmp[c];
            row[c] += term[c];
        }
    }

    // Scatter into WMMA B-fragment layout.
    const int h  = tIdx & 1;     // which half of the N=16 columns (t0 vs t1)
    const int tp = tIdx >> 1;    // t-pair index
#pragma unroll
    for (int ks = 0; ks < 2; ++ks) {
        const int base = (tp * 2 + ks) * 32;
        v2f lo, hi;
        lo.x = row[4 * ks + 0]; lo.y = row[4 * ks + 1];
        hi.x = row[4 * ks + 2]; hi.y = row[4 * ks + 3];
        epack[base + 8 * h + i]      = lo;   // lanes 0..15  (K0 = 0)
        epack[base + 16 + 8 * h + i] = hi;   // lanes 16..31 (K0 = 2)
    }
}

// ---------------------------------------------------------------------------
// Phase 2: out-tile = X(16x8) @ [E[t0]^T | E[t1]^T] (8x16) via two chained
// V_WMMA_F32_16X16X4_F32 (K split 0..3 / 4..7). Wave keeps its A fragment
// (x rows) in registers across 32 t-pairs; results transposed via a
// WAVE-PRIVATE LDS tile (DS ops from one wave execute in order, so only a
// compiler ordering fence is needed — no block barrier, no storecnt stall),
// then emitted as two contiguous 512-byte non-temporal b128 stores.
// ---------------------------------------------------------------------------
__global__ __launch_bounds__(256) void flow_wmma_kernel(
    const float* __restrict__ x, const v2f* __restrict__ epack,
    float* __restrict__ out)
{
    __shared__ __align__(16) float lds[8][16 * LDS_STRIDE];

    const int lane = threadIdx.x & 31;
    const int wave = threadIdx.x >> 5;
    const int half = lane >> 4;    // selects K0 = 2*half in A/B fragments
    const int col  = lane & 15;    // M index for A rows / N index for D cols

    const int ntile = blockIdx.x * 8 + wave;   // 128 * 8 = 1024 tiles of 16 rows
    const int nbase = ntile * 16;
    const int tp0   = blockIdx.y * 32;         // 8 * 32 = 256 t-pairs

    // A fragment: lane holds x[nbase+col, K0], x[.., K0+1] for each K-step.
    const v2f* xr = (const v2f*)(x + (size_t)(nbase + col) * 8);
    const v2f a0 = xr[half];       // j = 2*half, 2*half+1
    const v2f a1 = xr[half + 2];   // j = 2*half+4, 2*half+5

    float* ldsw = &lds[wave][0];
    const int nloc = lane >> 1;          // readback row (0..15)
    const int c4   = (lane & 1) * 4;     // readback col base (0 or 4)

    // Software-pipelined B-fragment loads (L2-resident, 512B/wave/iter).
    v2f b0 = epack[(tp0 * 2 + 0) * 32 + lane];
    v2f b1 = epack[(tp0 * 2 + 1) * 32 + lane];

    for (int tp = tp0; tp < tp0 + 32; ++tp) {
        const int tpn = (tp + 1 < tp0 + 32) ? (tp + 1) : tp;  // clamped, branchless
        const v2f nb0 = epack[(tpn * 2 + 0) * 32 + lane];
        const v2f nb1 = epack[(tpn * 2 + 1) * 32 + lane];

        v8f c = {};
        c = __builtin_amdgcn_wmma_f32_16x16x4_f32(false, a0, false, b0,
                                                  (short)0, c, false, false);
        c = __builtin_amdgcn_wmma_f32_16x16x4_f32(false, a1, false, b1,
                                                  (short)0, c, false, false);

        // D layout: VGPR r, lane l -> M = r + 8*half (local n), N = col.
#pragma unroll
        for (int r = 0; r < 8; ++r)
            ldsw[(r + 8 * half) * LDS_STRIDE + col] = c[r];
        __builtin_amdgcn_wave_barrier();   // order LDS write -> read (same wave)

        // cols 0..7 -> t0 block, cols 8..15 -> t1 block; each block is a
        // contiguous 512B run in memory (16 consecutive n-rows of 8 floats).
        const v4f v0 = *(const v4f*)&ldsw[nloc * LDS_STRIDE + c4];
        const v4f v1 = *(const v4f*)&ldsw[nloc * LDS_STRIDE + 8 + c4];
        __builtin_amdgcn_wave_barrier();   // order LDS read -> next write

        const size_t t0 = (size_t)tp * 2;
        float* o0 = out + (t0 * N_ROWS + nbase) * 8 + lane * 4;
        float* o1 = out + ((t0 + 1) * N_ROWS + nbase) * 8 + lane * 4;
        __builtin_nontemporal_store(v0, (v4f*)o0);
        __builtin_nontemporal_store(v1, (v4f*)o1);

        b0 = nb0;
        b1 = nb1;
    }
}

// ---------------------------------------------------------------------------
extern "C" void kernel_launch(void* const* d_in, const int* in_sizes, int n_in,
                              void* d_out, int out_size, void* d_ws, size_t ws_size,
                              hipStream_t stream)
{
    const float* x  = (const float*)d_in[0];   // [16384, 8]
    const float* t  = (const float*)d_in[1];   // [512]
    const float* M  = (const float*)d_in[2];   // [8, 8]
    const float* M0 = (const float*)d_in[3];   // [8, 8]
    float* out = (float*)d_out;                // [512, 16384, 8]
    v2f* epack = (v2f*)d_ws;                   // 256 * 2 * 32 * 8B = 128 KB

    expm_pack_kernel<<<16, 256, 0, stream>>>(t, M, M0, epack);

    dim3 grid(128, 8);
    flow_wmma_kernel<<<grid, 256, 0, stream>>>(x, epack, out);
}